// TabularLoss_28037546508514
// MI455X (gfx1250) — compile-verified
//
#include <hip/hip_runtime.h>
#include <hip/hip_bf16.h>
#include <math.h>

typedef __attribute__((ext_vector_type(2))) float v2f;
typedef __attribute__((ext_vector_type(4))) float v4f;
typedef __attribute__((ext_vector_type(8))) float v8f;

#define N_ROWS   131072
#define N_FEAT   512
#define DSAMPLE  8
#define NBATCH   (N_ROWS / DSAMPLE)   // 16384
#define NSHARED  128

// ---------- shared epilogue: per-wave energy from a 16x16 Gram in C-layout ----------
// C layout (32-bit 16x16): lane l holds column n = l&15; VGPR v holds row m = v + 8*(l>>4).
// Rows/cols 0-7 = x points, 8-15 = y points. energy = (sum_cross - sum_same)/128.
__device__ __forceinline__ float energy_from_gram(v8f c) {
    const int lane = threadIdx.x & 31;
    const int n = lane & 15;
    const int h = lane >> 4;
    // Diagonal holders: lanes 0-7 hold G[l][l] in c[l]; lanes 24-31 hold G[l-16][l-16] in c[l-24].
    int sel = ((h == 0) ? lane : (lane - 24)) & 7;
    float dv = c[0];
    dv = (sel == 1) ? c[1] : dv;
    dv = (sel == 2) ? c[2] : dv;
    dv = (sel == 3) ? c[3] : dv;
    dv = (sel == 4) ? c[4] : dv;
    dv = (sel == 5) ? c[5] : dv;
    dv = (sel == 6) ? c[6] : dv;
    dv = (sel == 7) ? c[7] : dv;
    // G[n][n]: held at lane n (n<8) or lane n+16 (n>=8)
    float dn = __shfl(dv, (n < 8) ? n : (n + 16), 32);
    float s = 0.0f;
#pragma unroll
    for (int v = 0; v < 8; ++v) {
        // m = v + 8*h ; G[m][m] held at lane v (h==0) or lane v+24 (h==1)
        float dm = __shfl(dv, (h == 0) ? v : (v + 24), 32);
        float sq = dm + dn - 2.0f * c[v];
        s += (sq > 0.0f) ? sqrtf(sq) : 0.0f;
    }
    // region of this lane's 8 entries: same-cloud iff h == (n>>3)
    bool same = (h == (n >> 3));
    s = same ? -s : s;
#pragma unroll
    for (int off = 16; off > 0; off >>= 1) s += __shfl_xor(s, off, 32);
    return s * (1.0f / 128.0f);
}

// ---------- kernel 0: zero workspace accumulators + gram buffer ----------
__global__ void tl_init_kernel(float* ws) {
    int i = blockIdx.x * blockDim.x + threadIdx.x;
    if (i < 2 + NSHARED * 256) ws[i] = 0.0f;
}

// ---------- kernel 1: gene term. One wave per batch; b128 loads feed 2 WMMAs each ----------
// K-permutation trick: columns are assigned {h=0: 8q..8q+3, h=1: 8q+4..8q+7}; since the
// identical register pair is passed as A and B, any column->slot assignment that is
// consistent per half yields the exact Gram.
__global__ __launch_bounds__(256) void tl_gene_kernel(const float* __restrict__ x,
                                                      const float* __restrict__ y,
                                                      float* __restrict__ acc_gene) {
    const int batch = blockIdx.x * 8 + (threadIdx.x >> 5);   // 2048 blocks * 8 waves = 16384
    const int lane  = threadIdx.x & 31;
    const int n = lane & 15;
    const int h = lane >> 4;
    const float* row = (n < 8)
        ? (x + (size_t)batch * (DSAMPLE * N_FEAT) + (size_t)n * N_FEAT)
        : (y + (size_t)batch * (DSAMPLE * N_FEAT) + (size_t)(n - 8) * N_FEAT);
    const float* p = row + 4 * h;

    v8f c = {};
    int q = 0;
    // columns [0,384): read once ever -> non-temporal
    for (; q < (N_FEAT - NSHARED) / 8; ++q) {
        v4f f = __builtin_nontemporal_load(reinterpret_cast<const v4f*>(p + 8 * q));
        v2f a0 = {f.x, f.y};
        v2f a1 = {f.z, f.w};
        c = __builtin_amdgcn_wmma_f32_16x16x4_f32(false, a0, false, a0, (short)0, c, false, false);
        c = __builtin_amdgcn_wmma_f32_16x16x4_f32(false, a1, false, a1, (short)0, c, false, false);
    }
    // tail columns [384,512): re-read by cell kernel -> keep cached
    for (; q < N_FEAT / 8; ++q) {
        v4f f = *reinterpret_cast<const v4f*>(p + 8 * q);
        v2f a0 = {f.x, f.y};
        v2f a1 = {f.z, f.w};
        c = __builtin_amdgcn_wmma_f32_16x16x4_f32(false, a0, false, a0, (short)0, c, false, false);
        c = __builtin_amdgcn_wmma_f32_16x16x4_f32(false, a1, false, a1, (short)0, c, false, false);
    }
    float e = energy_from_gram(c);
    if (lane == 0) atomicAdd(acc_gene, e);
}

// ---------- kernel 2: cell partial Grams. LDS-staged transpose, WMMA over batch axis ----------
#define BT        16                  // batch tile (K per tile)
#define LDK       28                  // row (m) stride in floats: 112B, 16B-aligned, 16 distinct banks
#define CS        452                 // channel stride in floats: mult of 4, == 4 (mod 64) banks
#define BATCH_PER_BLOCK 64            // 256 blocks cover 16384 batches
#define NTILES    (BATCH_PER_BLOCK / BT)   // 4

// k-position XOR swizzle (bits 2-3, per channel) to spread transpose-store banks.
__device__ __forceinline__ int kswz(int c) { return 4 * ((c >> 2) & 3); }

__global__ __launch_bounds__(256) void tl_cell_gram_kernel(const float* __restrict__ x,
                                                           const float* __restrict__ y,
                                                           float* __restrict__ gram) {
    __shared__ __align__(16) float sm[NSHARED * CS];          // 128*452*4 = 226 KB
    const int tid  = threadIdx.x;
    const int lane = tid & 31;
    const int wv   = tid >> 5;                // 0..7 -> channels [wv*16, wv*16+16)
    const int n = lane & 15;
    const int h = lane >> 4;
    const int bstart = blockIdx.x * BATCH_PER_BLOCK;

    v8f acc[16];
#pragma unroll
    for (int i = 0; i < 16; ++i) acc[i] = (v8f){};

    for (int t = 0; t < NTILES; ++t) {
        __syncthreads();
        const int b0 = bstart + t * BT;
        // stage: BT batches x 16 rows x 128 tail cols = 8192 float4s, coalesced global loads
#pragma unroll
        for (int qq = 0; qq < 32; ++qq) {
            int idx = qq * 256 + tid;
            int c4  = idx & 31;            // float4 group within the 128 tail cols
            int m   = (idx >> 5) & 15;     // row (0-7: x point, 8-15: y point)
            int bl  = idx >> 9;            // batch within tile
            const float* src = ((m < 8) ? x : y)
                + (size_t)(b0 + bl) * (DSAMPLE * N_FEAT)
                + (size_t)(m & 7) * N_FEAT + (N_FEAT - NSHARED) + c4 * 4;
            v4f f = *reinterpret_cast<const v4f*>(src);
#pragma unroll
            for (int j = 0; j < 4; ++j) {
                int cc = c4 * 4 + j;
                sm[cc * CS + m * LDK + (bl ^ kswz(cc))] = f[j];
            }
        }
        __syncthreads();
        // compute: per channel, one b128 LDS read feeds 2 WMMAs; 4 WMMAs cover BT=16
#pragma unroll
        for (int ci = 0; ci < 16; ++ci) {
            const int cc = wv * 16 + ci;
            const float* base = &sm[cc * CS + n * LDK];
            const int swz = kswz(cc);
#pragma unroll
            for (int g = 0; g < 2; ++g) {
                int koff = (8 * g + 4 * h) ^ swz;
                v4f f = *reinterpret_cast<const v4f*>(base + koff);
                v2f a0 = {f.x, f.y};
                v2f a1 = {f.z, f.w};
                acc[ci] = __builtin_amdgcn_wmma_f32_16x16x4_f32(false, a0, false, a0, (short)0,
                                                                acc[ci], false, false);
                acc[ci] = __builtin_amdgcn_wmma_f32_16x16x4_f32(false, a1, false, a1, (short)0,
                                                                acc[ci], false, false);
            }
        }
    }
    // merge partial Grams
#pragma unroll
    for (int ci = 0; ci < 16; ++ci) {
        int cc = wv * 16 + ci;
#pragma unroll
        for (int v = 0; v < 8; ++v) {
            int m = v + 8 * h;
            atomicAdd(&gram[cc * 256 + m * 16 + n], acc[ci][v]);
        }
    }
}

// ---------- kernel 3: cell epilogue. One wave per channel ----------
__global__ __launch_bounds__(256) void tl_cell_finish_kernel(const float* __restrict__ gram,
                                                             float* __restrict__ acc_cell) {
    const int c    = blockIdx.x * 8 + (threadIdx.x >> 5);    // 16 blocks * 8 waves = 128
    const int lane = threadIdx.x & 31;
    const int n = lane & 15;
    const int h = lane >> 4;
    v8f g;
#pragma unroll
    for (int v = 0; v < 8; ++v) g[v] = gram[c * 256 + (v + 8 * h) * 16 + n];
    float e = energy_from_gram(g);
    if (lane == 0) atomicAdd(acc_cell, e);
}

// ---------- kernel 4: combine ----------
__global__ void tl_final_kernel(const float* __restrict__ ws, float* __restrict__ out) {
    out[0] = ws[0] * (1.0f / (float)NBATCH) + ws[1] * (1.0f / (float)NSHARED);
}

extern "C" void kernel_launch(void* const* d_in, const int* in_sizes, int n_in,
                              void* d_out, int out_size, void* d_ws, size_t ws_size,
                              hipStream_t stream) {
    const float* x = (const float*)d_in[0];
    const float* y = (const float*)d_in[1];
    float* ws = (float*)d_ws;
    float* acc_gene = ws + 0;
    float* acc_cell = ws + 1;
    float* gram     = ws + 2;            // 128 * 256 floats

    tl_init_kernel<<<(2 + NSHARED * 256 + 255) / 256, 256, 0, stream>>>(ws);
    tl_gene_kernel<<<NBATCH / 8, 256, 0, stream>>>(x, y, acc_gene);
    tl_cell_gram_kernel<<<NBATCH / BATCH_PER_BLOCK, 256, 0, stream>>>(x, y, gram);
    tl_cell_finish_kernel<<<NSHARED / 8, 256, 0, stream>>>(gram, acc_cell);
    tl_final_kernel<<<1, 1, 0, stream>>>(ws, (float*)d_out);
}